// QuantumSubgenerator_25314537242888
// MI455X (gfx1250) — compile-verified
//
#include <hip/hip_runtime.h>

// Quantum sub-generator: 12 qubits, batch 1024, real amplitudes.
// One workgroup = 16 wave32s = 16 batch items; each wave keeps its 4096-float
// state vector in LDS (16 KB). 16*16KB + 1KB accumulator = 257 KB dynamic LDS,
// which fits CDNA5's 320 KB-per-workgroup LDS (impossible pre-CDNA5).
// Gates run barrier-free per wave (same-wave DS ops are in-order). The Z-expval
// readout out[b,n] = sum_k state[b,k]^2 * (bit_n(k) ? -1 : +1) is done as a
// real GEMM on the WMMA pipe: V_WMMA_F32_16X16X4_F32, M=16 batch, N=16 (12
// wires used), K=4096 split 256-wide across the 16 waves.
//
// Round-2 refinements (from round-1 asm): branchless +-1 sign synthesis via
// sign-bit bit ops (kills v_cmp/v_cndmask chains next to each WMMA; B.y is a
// single XOR since k0 is even), and float2 A-loads (ds_load_b64).

#define NQ   12
#define DIM  4096      // 2^12 amplitudes
#define BPW  16        // batch items (== waves) per workgroup
#define WSZ  32        // wave32

typedef __attribute__((ext_vector_type(2))) float v2f;
typedef __attribute__((ext_vector_type(8))) float v8f;

__device__ __forceinline__ void apply_ry(float* __restrict__ S, int q, float theta, int lane) {
    float sn, cs;
    __sincosf(0.5f * theta, &sn, &cs);
    const int post = 1 << (11 - q);      // pair stride; wire q <-> index bit (11-q)
    const int pm   = post - 1;
    #pragma unroll 4
    for (int p = lane; p < DIM / 2; p += WSZ) {
        int lo = p & pm;                 // bits below the gate bit
        int i0 = ((p ^ lo) << 1) | lo;   // insert 0 at the gate bit position
        int i1 = i0 + post;              // gate bit = 1
        float s0 = S[i0], s1 = S[i1];
        S[i0] = cs * s0 - sn * s1;
        S[i1] = sn * s0 + cs * s1;
    }
}

__device__ __forceinline__ void apply_cnot(float* __restrict__ S, int cwire, int twire, int lane) {
    const int cb = 11 - cwire, tb = 11 - twire;   // bit positions
    const int p1 = cb < tb ? cb : tb;
    const int p2 = cb < tb ? tb : cb;
    const int m1 = (1 << p1) - 1;
    const int gs = p2 - p1 - 1;                   // gap size between the two holes
    const int mg = (1 << gs) - 1;
    #pragma unroll 4
    for (int j = lane; j < DIM / 4; j += WSZ) {   // enumerate ctrl=1, tgt=0 indices
        int low  = j & m1;
        int rest = j >> p1;
        int mid  = rest & mg;
        int high = rest >> gs;
        int i  = (high << (p2 + 1)) | (mid << (p1 + 1)) | low | (1 << cb);
        int i2 = i | (1 << tb);
        float a = S[i], b = S[i2];                // swap tgt=0 <-> tgt=1
        S[i]  = b;
        S[i2] = a;
    }
}

__global__ __launch_bounds__(BPW * WSZ)
void qsim_wmma_kernel(const float* __restrict__ latent,   // (1024, 12)
                      const float* __restrict__ weights,  // (2, 12)
                      float* __restrict__ out)            // (1024, 12)
{
    extern __shared__ float smem[];               // [BPW*DIM] states | [256] acc
    const int tid  = threadIdx.x;
    const int wave = tid >> 5;
    const int lane = tid & 31;
    const int b    = blockIdx.x * BPW + wave;

    float* S   = smem + wave * DIM;
    float* acc = smem + BPW * DIM;

    // |0...0> init (own-wave LDS only; in-order DS => no barrier needed)
    for (int k = lane; k < DIM; k += WSZ) S[k] = 0.0f;
    if (lane == 0) S[0] = 1.0f;
    if (tid < 256) acc[tid] = 0.0f;

    // Layer 1: per-batch RY(latent)
    for (int q = 0; q < NQ; ++q) apply_ry(S, q, latent[b * NQ + q], lane);
    // Layer 2: shared RY(weights[0])
    for (int q = 0; q < NQ; ++q) apply_ry(S, q, weights[q], lane);
    // Entangler: CNOT chain + wrap
    for (int q = 0; q < NQ - 1; ++q) apply_cnot(S, q, q + 1, lane);
    apply_cnot(S, NQ - 1, 0, lane);
    // Layer 3: shared RY(weights[1])
    for (int q = 0; q < NQ; ++q) apply_ry(S, q, weights[NQ + q], lane);

    __syncthreads();   // every wave now reads all 16 states for the GEMM

    // --- Z-expectation GEMM on the WMMA pipe ---
    // D = A x B + C with A(16x4)=probs[m][k], B(4x16)=sign(k,n), C(16x16) f32.
    // A layout: lane(0..15)+16h holds M=lane&15, K={k0, k0+1} in 2 VGPRs.
    // B layout: lane holds N=lane&15, rows K={k0, k0+1} in 2 VGPRs.
    const int m = lane & 15;                  // M for A, N for B (same lane slot)
    const int h = lane >> 4;
    const float* Sm = smem + m * DIM;         // batch m's state
    const int kbase = wave * (DIM / BPW);     // this wave's 256-wide K chunk

    // Hoisted lane constants for branchless sign generation:
    //   sign(k, n) = as_float(0x3F800000 | (bit_{11-n}(k) << 31))  (+-1.0f)
    // Column n's bit sits at position 11-n, so shifting k left by (20+n) puts
    // it in the float sign bit. Dead columns (n >= 12) get base=mask=0 -> 0.0f.
    const bool     live = (m < NQ);
    const unsigned base = live ? 0x3F800000u : 0u;
    const unsigned smsk = live ? 0x80000000u : 0u;
    const int      sha  = live ? (20 + m)    : 0;
    // k0 is always even; k0+1 differs only in bit 0 == column n=11's bit.
    const unsigned yxor = (m == NQ - 1) ? 0x80000000u : 0u;

    v8f C = {};
    for (int it = 0; it < (DIM / BPW) / 4; ++it) {
        const int k0 = kbase + it * 4 + 2 * h;
        const float2 a = *(const float2*)(Sm + k0);   // aligned: k0 even
        v2f A; A.x = a.x * a.x; A.y = a.y * a.y;      // probabilities
        const unsigned bx = base | (((unsigned)k0 << sha) & smsk);
        v2f B;
        B.x = __uint_as_float(bx);
        B.y = __uint_as_float(bx ^ yxor);
        C = __builtin_amdgcn_wmma_f32_16x16x4_f32(
                /*neg_a=*/false, A, /*neg_b=*/false, B,
                /*c_mod=*/(short)0, C, /*reuse_a=*/false, /*reuse_b=*/false);
    }

    // Reduce the 16 per-wave K-partials: ds_add_f32 into acc[16][16].
    // C/D layout: VGPR j of lane -> (M = j + 8h, N = lane&15).
    #pragma unroll
    for (int j = 0; j < 8; ++j)
        atomicAdd(&acc[(j + 8 * h) * 16 + m], C[j]);

    __syncthreads();

    if (tid < BPW * NQ) {
        int mm = tid / NQ, nn = tid % NQ;
        out[(blockIdx.x * BPW + mm) * NQ + nn] = acc[mm * 16 + nn];
    }
}

extern "C" void kernel_launch(void* const* d_in, const int* in_sizes, int n_in,
                              void* d_out, int out_size, void* d_ws, size_t ws_size,
                              hipStream_t stream) {
    (void)in_sizes; (void)n_in; (void)d_ws; (void)ws_size; (void)out_size;
    const float* latent  = (const float*)d_in[0];   // (1024, 12) f32
    const float* weights = (const float*)d_in[1];   // (2, 12) f32
    float* out = (float*)d_out;                     // (1024, 12) f32

    const size_t lds_bytes = (size_t)(BPW * DIM + 256) * sizeof(float); // 257 KB of 320 KB
    qsim_wmma_kernel<<<dim3(1024 / BPW), dim3(BPW * WSZ), lds_bytes, stream>>>(
        latent, weights, out);
}